// Filter_encoder_86217173500043
// MI455X (gfx1250) — compile-verified
//
#include <hip/hip_runtime.h>
#include <hip/hip_bf16.h>
#include <math.h>

// ---------------------------------------------------------------------------
// Filter_encoder (ChebConv K=25 -> LeakyReLU -> mu/logstd heads) for gfx1250
// f32 end-to-end; dense GEMMs via V_WMMA_F32_16X16X4_F32 (wave32 WMMA).
// W pre-packed into WMMA-native pair layout -> 1x global_load_b64 per B step.
// ---------------------------------------------------------------------------

#define N_NODES 50000
#define N_EDGES 800000
#define IN_CH   128
#define HID_CH  128
#define OUT_CH  64
#define K_ORD   25

typedef float v2f __attribute__((ext_vector_type(2)));
typedef float v8f __attribute__((ext_vector_type(8)));

// ---------------- utility ----------------
__global__ void fill4_kernel(float4* __restrict__ p, float v, int n4) {
    int i = blockIdx.x * blockDim.x + threadIdx.x;
    if (i < n4) p[i] = make_float4(v, v, v, v);
}

__global__ void fill_kernel(float* __restrict__ p, float v, int n) {
    int i = blockIdx.x * blockDim.x + threadIdx.x;
    if (i < n) p[i] = v;
}

__global__ void deg_kernel(const int* __restrict__ src, float* __restrict__ deg, int nE) {
    int e = blockIdx.x * blockDim.x + threadIdx.x;
    if (e < nE) atomicAdd(&deg[src[e]], 1.0f);
}

__global__ void dis_kernel(float* __restrict__ deg, int n) {
    int i = blockIdx.x * blockDim.x + threadIdx.x;
    if (i < n) {
        float d = deg[i];
        deg[i] = (d > 0.0f) ? rsqrtf(d) : 0.0f;   // deg buffer becomes dis
    }
}

__global__ void enorm_kernel(const int* __restrict__ src, const int* __restrict__ dst,
                             const float* __restrict__ dis, float* __restrict__ en, int nE) {
    int e = blockIdx.x * blockDim.x + threadIdx.x;
    if (e < nE) en[e] = -dis[src[e]] * dis[dst[e]];
}

// ---------------- weight pre-pack into WMMA-native pair layout -------------
// P[((k4*ncols + c)*2 + h)*2 + {0,1}] = W[(4*k4+2h+{0,1})*ncols + c]
// => lane (half,r) of N-tile n0 loads its (b.x,b.y) as one contiguous b64,
//    and a full wave's k-step footprint is one contiguous 256B line.
__global__ void packw_kernel(const float* __restrict__ W, float* __restrict__ P, int ncols) {
    int idx = blockIdx.x * blockDim.x + threadIdx.x;   // over 32 * ncols * 2
    if (idx >= 32 * ncols * 2) return;
    int h  = idx & 1;
    int c  = (idx >> 1) % ncols;
    int k4 = (idx >> 1) / ncols;
    P[2 * idx + 0] = W[(size_t)(4 * k4 + 2 * h + 0) * ncols + c];
    P[2 * idx + 1] = W[(size_t)(4 * k4 + 2 * h + 1) * ncols + c];
}

// ---------------- sparse propagate: p[dst] += enorm * h[src] ----------------
// one wave per edge; indices forced uniform -> scalar (s_load) path.
__global__ void prop_kernel(const float* __restrict__ h,
                            const int* __restrict__ src, const int* __restrict__ dst,
                            const float* __restrict__ en, float* __restrict__ p, int nE) {
    int gtid = blockIdx.x * blockDim.x + threadIdx.x;
    int e    = gtid >> 5;
    if (e >= nE) return;
    e = __builtin_amdgcn_readfirstlane(e);    // wave-uniform: use SMEM path
    int lane = gtid & 31;
    int s = src[e];
    int d = dst[e];
    float w = en[e];
    float4 hv = ((const float4*)(h + (size_t)s * HID_CH))[lane];
    float* pd = p + (size_t)d * HID_CH + lane * 4;
    atomicAdd(pd + 0, w * hv.x);
    atomicAdd(pd + 1, w * hv.y);
    atomicAdd(pd + 2, w * hv.z);
    atomicAdd(pd + 3, w * hv.w);
}

// ---------------- Chebyshev recurrence: p = 2*p - prev ----------------
__global__ void cheb4_kernel(float4* __restrict__ p, const float4* __restrict__ prev, int n4) {
    int i = blockIdx.x * blockDim.x + threadIdx.x;
    if (i < n4) {
        float4 a = p[i], b = prev[i];
        p[i] = make_float4(2.0f * a.x - b.x, 2.0f * a.y - b.y,
                           2.0f * a.z - b.z, 2.0f * a.w - b.w);
    }
}

// ---------------- dense: ACC(50000x128) += T(50000x128) @ Wk(128x128) ------
__global__ void gemm_acc_kernel(const float* __restrict__ T, const float* __restrict__ Pk,
                                float* __restrict__ ACC) {
    __shared__ float sA[16 * 132];
    const int row0 = blockIdx.x * 16;
    const int tid  = threadIdx.x;

    if (tid < 128) __builtin_prefetch(Pk + tid * 128, 0, 1);

    for (int i = tid; i < 16 * HID_CH; i += 256)
        sA[(i >> 7) * 132 + (i & 127)] = T[(size_t)row0 * HID_CH + i];
    __syncthreads();

    const int wave = tid >> 5;
    const int lane = tid & 31;
    const int half = lane >> 4;   // 0: lanes 0-15, 1: lanes 16-31
    const int r    = lane & 15;
    const int n0   = wave * 16;
    const float2* Bp = (const float2*)Pk;

    // C/D 16x16 f32 layout: VGPR i -> row (i + 8*half), col (n0 + r)
    v8f c;
#pragma unroll
    for (int i = 0; i < 8; ++i)
        c[i] = ACC[(size_t)(row0 + i + 8 * half) * HID_CH + n0 + r];

#pragma unroll
    for (int k4 = 0; k4 < 32; ++k4) {
        v2f a, b;
        a.x = sA[r * 132 + 4 * k4 + 2 * half];
        a.y = sA[r * 132 + 4 * k4 + 2 * half + 1];
        float2 bb = Bp[(k4 * 128 + n0 + r) * 2 + half];
        b.x = bb.x;
        b.y = bb.y;
        c = __builtin_amdgcn_wmma_f32_16x16x4_f32(false, a, false, b,
                                                  (short)0, c, false, false);
    }

#pragma unroll
    for (int i = 0; i < 8; ++i)
        ACC[(size_t)(row0 + i + 8 * half) * HID_CH + n0 + r] = c[i];
}

// ---------------- fused head: bias + LeakyReLU + mu/ls GEMM + normalize ----
__global__ void head_kernel(const float* __restrict__ ACC, const float* __restrict__ bias,
                            const float* __restrict__ Pmu, const float* __restrict__ b_mu,
                            const float* __restrict__ Pls, const float* __restrict__ b_ls,
                            float* __restrict__ out) {
    __shared__ float sH[16 * 132];
    __shared__ float sMu[16 * 64];
    __shared__ float sLs[16 * 64];
    __shared__ float sNrm[16];

    const int row0 = blockIdx.x * 16;
    const int tid  = threadIdx.x;

    for (int i = tid; i < 16 * HID_CH; i += 256) {
        float h = ACC[(size_t)row0 * HID_CH + i] + bias[i & 127];
        sH[(i >> 7) * 132 + (i & 127)] = (h > 0.0f) ? h : 0.01f * h;  // LeakyReLU
    }
    __syncthreads();

    const int wave = tid >> 5;
    const int lane = tid & 31;
    const int half = lane >> 4;
    const int r    = lane & 15;
    const int hsel = wave >> 2;          // 0 = mu head, 1 = ls head
    const int n0   = (wave & 3) * 16;
    const float2* Bp = (const float2*)(hsel ? Pls : Pmu);
    const float*  bh = hsel ? b_ls : b_mu;

    v8f c;
    float cb = bh[n0 + r];               // bias broadcast down the column
#pragma unroll
    for (int i = 0; i < 8; ++i) c[i] = cb;

#pragma unroll
    for (int k4 = 0; k4 < 32; ++k4) {
        v2f a, b;
        a.x = sH[r * 132 + 4 * k4 + 2 * half];
        a.y = sH[r * 132 + 4 * k4 + 2 * half + 1];
        float2 bb = Bp[(k4 * 64 + n0 + r) * 2 + half];
        b.x = bb.x;
        b.y = bb.y;
        c = __builtin_amdgcn_wmma_f32_16x16x4_f32(false, a, false, b,
                                                  (short)0, c, false, false);
    }

    float* dstL = hsel ? sLs : sMu;
#pragma unroll
    for (int i = 0; i < 8; ++i)
        dstL[(i + 8 * half) * OUT_CH + n0 + r] = c[i];
    __syncthreads();

    if (tid < 16) {
        float s = 0.0f;
        for (int n = 0; n < OUT_CH; ++n) { float v = sLs[tid * OUT_CH + n]; s += v * v; }
        sNrm[tid] = sqrtf(s);
    }
    __syncthreads();

    const size_t ls_off   = (size_t)N_NODES * OUT_CH;
    const size_t zeta_off = 2 * ls_off;
    for (int idx = tid; idx < 16 * OUT_CH; idx += 256) {
        int row = idx >> 6, col = idx & 63;
        size_t o = (size_t)(row0 + row) * OUT_CH + col;
        float m = sMu[idx];
        out[o]            = m;                 // mu
        out[zeta_off + o] = m;                 // zeta (eval-mode reparam == mu)
        float l = sLs[idx];
        out[ls_off + o]   = l / fmaxf(sNrm[row], 1e-12f) * 1.8f;  // logstd
    }
}

// ---------------------------------------------------------------------------
extern "C" void kernel_launch(void* const* d_in, const int* in_sizes, int n_in,
                              void* d_out, int out_size, void* d_ws, size_t ws_size,
                              hipStream_t stream) {
    const float* x    = (const float*)d_in[0];
    const int*   ei   = (const int*)  d_in[1];
    const float* W    = (const float*)d_in[2];   // [K,128,128]
    const float* b    = (const float*)d_in[3];
    const float* w_mu = (const float*)d_in[4];
    const float* b_mu = (const float*)d_in[5];
    const float* w_ls = (const float*)d_in[6];
    const float* b_ls = (const float*)d_in[7];
    float* out = (float*)d_out;

    const int N = N_NODES, E = N_EDGES;
    const int NC = N * HID_CH;                   // 6.4M floats per feature buffer
    const int* src = ei;
    const int* dst = ei + E;

    // workspace carve-up (~108 MB total)
    float* ws   = (float*)d_ws;
    float* deg  = ws;                            // N   (becomes dis)
    float* en   = deg + N;                       // E
    float* B0   = en + E;                        // N*128
    float* B1   = B0 + (size_t)NC;               // N*128
    float* B2   = B1 + (size_t)NC;               // N*128
    float* acc  = B2 + (size_t)NC;               // N*128
    float* PW   = acc + (size_t)NC;              // 25 * 128*128 packed weights
    float* Pmu  = PW + (size_t)K_ORD * IN_CH * HID_CH;  // 128*64
    float* Pls  = Pmu + (size_t)IN_CH * OUT_CH;         // 128*64

    const int TB = 256;
    const int gN   = (N + TB - 1) / TB;
    const int gE   = (E + TB - 1) / TB;
    const int gNC4 = (NC / 4 + TB - 1) / TB;     // float4 granularity
    const int gP   = (E * 32 + TB - 1) / TB;     // one wave per edge
    const int gG   = N / 16;                     // 3125 row-tiles

    // --- pack weights into WMMA-native layout (tiny, once per launch) ---
    for (int k = 0; k < K_ORD; ++k)
        packw_kernel<<<(32 * 128 * 2 + TB - 1) / TB, TB, 0, stream>>>(
            W + (size_t)k * IN_CH * HID_CH, PW + (size_t)k * IN_CH * HID_CH, HID_CH);
    packw_kernel<<<(32 * 64 * 2 + TB - 1) / TB, TB, 0, stream>>>(w_mu, Pmu, OUT_CH);
    packw_kernel<<<(32 * 64 * 2 + TB - 1) / TB, TB, 0, stream>>>(w_ls, Pls, OUT_CH);

    // --- edge normalization ---
    fill_kernel<<<gN, TB, 0, stream>>>(deg, 0.0f, N);
    deg_kernel<<<gE, TB, 0, stream>>>(src, deg, E);
    dis_kernel<<<gN, TB, 0, stream>>>(deg, N);
    enorm_kernel<<<gE, TB, 0, stream>>>(src, dst, deg, en, E);

    // --- Chebyshev order 0/1 ---
    fill4_kernel<<<gNC4, TB, 0, stream>>>((float4*)acc, 0.0f, NC / 4);
    fill4_kernel<<<gNC4, TB, 0, stream>>>((float4*)B1, 0.0f, NC / 4);
    prop_kernel<<<gP, TB, 0, stream>>>(x, src, dst, en, B1, E);     // t1 = prop(x)
    gemm_acc_kernel<<<gG, TB, 0, stream>>>(x,  PW,                          acc); // += t0 @ W0
    gemm_acc_kernel<<<gG, TB, 0, stream>>>(B1, PW + (size_t)IN_CH * HID_CH, acc); // += t1 @ W1

    // --- Chebyshev orders 2..24 (3-buffer rotation) ---
    float* bufs[3] = {B0, B1, B2};
    const float* prev = x;      // t_{k-2}
    float* curr = B1;           // t_{k-1}
    int fi = 2;                 // next free buffer index
    for (int k = 2; k < K_ORD; ++k) {
        float* fr = bufs[fi];
        fill4_kernel<<<gNC4, TB, 0, stream>>>((float4*)fr, 0.0f, NC / 4);
        prop_kernel<<<gP, TB, 0, stream>>>(curr, src, dst, en, fr, E);         // fr = prop(t_{k-1})
        cheb4_kernel<<<gNC4, TB, 0, stream>>>((float4*)fr, (const float4*)prev, NC / 4);
        gemm_acc_kernel<<<gG, TB, 0, stream>>>(fr, PW + (size_t)k * IN_CH * HID_CH, acc);
        prev = curr;
        curr = fr;
        fi = (fi + 1) % 3;
    }

    // --- fused bias + LeakyReLU + heads + normalize (packed head weights) ---
    head_kernel<<<gG, TB, 0, stream>>>(acc, b, Pmu, b_mu, Pls, b_ls, out);
}